// DependencyLinearLayer_39779987096218
// MI455X (gfx1250) — compile-verified
//
#include <hip/hip_runtime.h>

// Problem constants (match reference)
#define B_    4
#define L_    256
#define IN_   768
#define H_    256
#define C_    12
#define NDEP_ 48

typedef __attribute__((ext_vector_type(2))) float v2f;
typedef __attribute__((ext_vector_type(8))) float v8f;

// Workspace layout (float offsets). Total 57952 floats = ~232 KB.
#define OFF_WS    0        // W_s_comb padded (768 x 16)
#define OFF_WT    12288    // W_t_comb padded (768 x 16)
#define OFF_DEP   24576    // dep_log + cls_b   (48 x 12)
#define OFF_BS    25152    // bias_s padded (16)  [bias_t at +16]
#define OFF_SLOG  25184    // s_log padded (1024 x 16)
#define OFF_TLOG  41568    // t_log padded (1024 x 16)

// ---------------------------------------------------------------------------
// Stage A: fold weights.
//   W_s_comb[k][c] = sum_h s_fc_w[h][k] * cls_w[c][h]          (768x16, pad 0)
//   W_t_comb[k][c] = sum_h t_fc_w[h][k] * cls_w[c][H+h]
//   dep_logc[n][c] = cls_b[c] + sum_h dep_emb[n][h]*cls_w[c][2H+h]   (48x12)
//   bias_s[c]      = sum_h s_fc_b[h] * cls_w[c][h]   (padded 16; bias_t next)
// Tiny (~5 MFLOP) -> plain VALU.
// ---------------------------------------------------------------------------
__global__ void __launch_bounds__(256)
fold_kernel(const float* __restrict__ sfw, const float* __restrict__ sfb,
            const float* __restrict__ tfw, const float* __restrict__ tfb,
            const float* __restrict__ dep_emb, const float* __restrict__ cls_w,
            const float* __restrict__ cls_b, float* __restrict__ ws) {
  int t = blockIdx.x * blockDim.x + threadIdx.x;
  if (t < 2 * 12288) {                       // combined weights
    int sel = (t >= 12288) ? 1 : 0;
    int u = t - sel * 12288;
    int k = u >> 4, n = u & 15;
    float acc = 0.f;
    if (n < C_) {
      const float* fw = sel ? tfw : sfw;
      const float* wc = cls_w + n * (3 * H_) + sel * H_;
      for (int h = 0; h < H_; ++h) acc += fw[h * IN_ + k] * wc[h];
    }
    ws[(sel ? OFF_WT : OFF_WS) + u] = acc;
  } else if (t < 24576 + NDEP_ * C_) {       // dep_log table (+cls_b fold)
    int u = t - 24576;
    int n = u / C_, c = u % C_;
    const float* wd = cls_w + c * (3 * H_) + 2 * H_;
    const float* de = dep_emb + n * H_;
    float acc = cls_b[c];
    for (int h = 0; h < H_; ++h) acc += de[h] * wd[h];
    ws[OFF_DEP + u] = acc;
  } else if (t < 24576 + NDEP_ * C_ + 32) {  // folded biases (s then t)
    int u = t - (24576 + NDEP_ * C_);
    int sel = (u >= 16) ? 1 : 0;
    int n = u & 15;
    float acc = 0.f;
    if (n < C_) {
      const float* fb = sel ? tfb : sfb;
      const float* wc = cls_w + n * (3 * H_) + sel * H_;
      for (int h = 0; h < H_; ++h) acc += fb[h] * wc[h];
    }
    ws[OFF_BS + u] = acc;
  }
}

// ---------------------------------------------------------------------------
// Stage B: fp32 WMMA projections.  s_log / t_log = x @ W_comb + bias.
// One wave (32 threads, EXEC all ones) per 16x16 output tile.
// A tile (16x32 chunk of x) staged in LDS with stride-33 padding.
// K loop: 192 x V_WMMA_F32_16X16X4_F32 per tile (full fp32 precision).
// A layout (ISA 7.12.2): lanes 0-15 -> M=lane, v0=K0,v1=K1; lanes16-31 K2/K3.
// B layout mirrors A; C/D: VGPR r holds rows r (lanes 0-15) and r+8 (16-31).
// ---------------------------------------------------------------------------
__global__ void __launch_bounds__(32)
proj_wmma_kernel(const float* __restrict__ x, float* __restrict__ ws) {
  __shared__ float lds_a[16 * 33];
  const int lane = threadIdx.x;
  const int half = lane >> 4;     // 0 or 1
  const int m    = lane & 15;
  const int row0 = blockIdx.x * 16;     // 64 tiles over 1024 rows
  const int sel  = blockIdx.y;          // 0 = s-stream, 1 = t-stream

  const float* __restrict__ W    = ws + (sel ? OFF_WT : OFF_WS);
  const float* __restrict__ bias = ws + OFF_BS + sel * 16;
  float* __restrict__ outlog     = ws + (sel ? OFF_TLOG : OFF_SLOG);

  v8f acc = {};
  for (int k0 = 0; k0 < IN_; k0 += 32) {
    // coalesced 16x32 chunk of x into LDS (one 128B line per row)
    #pragma unroll
    for (int r = 0; r < 16; ++r)
      lds_a[r * 33 + lane] = x[(row0 + r) * IN_ + k0 + lane];
    __syncthreads();
    #pragma unroll
    for (int kk = 0; kk < 32; kk += 4) {
      const int ka = kk + 2 * half;
      v2f a, b;
      a.x = lds_a[m * 33 + ka];
      a.y = lds_a[m * 33 + ka + 1];
      b.x = W[(k0 + ka) * 16 + m];
      b.y = W[(k0 + ka + 1) * 16 + m];
      acc = __builtin_amdgcn_wmma_f32_16x16x4_f32(
          /*neg_a=*/false, a, /*neg_b=*/false, b,
          /*c_mod=*/(short)0, acc, /*reuse_a=*/false, /*reuse_b=*/false);
    }
    __syncthreads();
  }
  const float bval = bias[m];
  #pragma unroll
  for (int r = 0; r < 8; ++r) {
    const int row = row0 + r + 8 * half;   // C/D VGPR layout
    outlog[row * 16 + m] = acc[r] + bval;
  }
}

// ---------------------------------------------------------------------------
// Stage C: bandwidth kernel.
//   out[b,i,c,j] = dep_logc[graph[b,i,j]][c] + s_log[b,i,c] + t_log[b,j,c]
// One block per (b,i); t_log row-block + 48x12 table + s_log row in LDS.
// Stores: 12 fully-coalesced 1 KB lines per block. ~14 MB total traffic.
// ---------------------------------------------------------------------------
__global__ void __launch_bounds__(256)
assemble_kernel(const int* __restrict__ graph, const float* __restrict__ ws,
                float* __restrict__ out) {
  __shared__ float t_lds[256 * 13];        // stride 13 -> conflict-free
  __shared__ float dep_lds[NDEP_ * C_];
  __shared__ float sc[C_];
  const int bi  = blockIdx.x;              // b*L + i
  const int b   = bi >> 8;                 // L == 256
  const int tid = threadIdx.x;

  if (tid < C_) sc[tid] = ws[OFF_SLOG + bi * 16 + tid];
  for (int idx = tid; idx < NDEP_ * C_; idx += 256)
    dep_lds[idx] = ws[OFF_DEP + idx];
  for (int idx = tid; idx < 256 * 16; idx += 256) {   // coalesced t_log block
    const int j = idx >> 4, c = idx & 15;
    const float v = ws[OFF_TLOG + (b * L_ + j) * 16 + c];
    if (c < C_) t_lds[j * 13 + c] = v;
  }
  __syncthreads();

  const int j = tid;
  const int g = graph[bi * L_ + j];
  const float* __restrict__ dl = dep_lds + g * C_;
  float tv[C_];
  #pragma unroll
  for (int c = 0; c < C_; ++c) tv[c] = t_lds[j * 13 + c];
  float* __restrict__ outp = out + (size_t)bi * C_ * L_ + j;
  #pragma unroll
  for (int c = 0; c < C_; ++c)
    outp[c * L_] = dl[c] + sc[c] + tv[c];
}

// ---------------------------------------------------------------------------
extern "C" void kernel_launch(void* const* d_in, const int* in_sizes, int n_in,
                              void* d_out, int out_size, void* d_ws, size_t ws_size,
                              hipStream_t stream) {
  (void)in_sizes; (void)n_in; (void)out_size; (void)ws_size;
  const float* x       = (const float*)d_in[0];   // (B,L,IN)
  const int*   graph   = (const int*)  d_in[1];   // (B,L,L)
  const float* sfw     = (const float*)d_in[2];   // (H,IN)
  const float* sfb     = (const float*)d_in[3];   // (H)
  const float* tfw     = (const float*)d_in[4];   // (H,IN)
  const float* tfb     = (const float*)d_in[5];   // (H)
  const float* dep_emb = (const float*)d_in[6];   // (NDEP,H)
  const float* cls_w   = (const float*)d_in[7];   // (C,3H)
  const float* cls_b   = (const float*)d_in[8];   // (C)
  float* out = (float*)d_out;
  float* ws  = (float*)d_ws;

  // Stage A: fold weights (25184 tasks)
  fold_kernel<<<99, 256, 0, stream>>>(sfw, sfb, tfw, tfb, dep_emb,
                                      cls_w, cls_b, ws);
  // Stage B: fp32 WMMA projections, 64 row-tiles x {s,t}
  proj_wmma_kernel<<<dim3(64, 2), 32, 0, stream>>>(x, ws);
  // Stage C: gather + broadcast-add, one block per (b,i)
  assemble_kernel<<<B_ * L_, 256, 0, stream>>>(graph, ws, out);
}